// Block_67147518706214
// MI455X (gfx1250) — compile-verified
//
#include <hip/hip_runtime.h>
#include <math.h>

#define DIM  768
#define NTOK 196
#define BATCH 64
#define ROWS (BATCH * NTOK)   // 12544
#define MLPD (4 * DIM)        // 3072
#define HID  64

typedef float v2f __attribute__((ext_vector_type(2)));
typedef float v8f __attribute__((ext_vector_type(8)));
typedef unsigned int u32x4 __attribute__((ext_vector_type(4)));
typedef int i32x4 __attribute__((ext_vector_type(4)));
typedef int i32x8 __attribute__((ext_vector_type(8)));

__device__ __forceinline__ float gelu_exact(float v) {
    return 0.5f * v * (1.0f + erff(v * 0.70710678118654752f));
}

// ---- Tensor Data Mover: load a 2-D fp32 tile (tile_w x tile_h) global->LDS,
// ---- padding +1 DWORD every 32 DWORDs so LDS row stride = 33 floats.
__device__ __forceinline__ void tdm_load_tile_f32(
        unsigned lds_off, const void* gptr, unsigned row_stride_elts,
        unsigned tile_w, unsigned tile_h)
{
    const unsigned long long ga = (unsigned long long)(uintptr_t)gptr;
    const unsigned td0 = 1u << 20, td1 = 1u << 20;   // tensor dims (tiles always in-bounds)
    u32x4 g0 = {
        1u,                                                 // count=1 valid, user mode
        lds_off,                                            // LDS byte address
        (unsigned)(ga & 0xFFFFFFFFu),                       // global_addr[31:0]
        (unsigned)((ga >> 32) & 0x01FFFFFFu) | (2u << 30)   // global_addr[56:32] | type=2
    };
    i32x8 g1 = {
        (int)((2u << 16) | (1u << 20) | (4u << 22)),        // data_size=4B, pad_en, every 32 DW, +1 DW
        (int)((td0 & 0xFFFFu) << 16),                       // abar=0 | tensor_dim0.lo
        (int)(((td0 >> 16) & 0xFFFFu) | ((td1 & 0xFFFFu) << 16)),
        (int)(((td1 >> 16) & 0xFFFFu) | ((tile_w & 0xFFFFu) << 16)),  // tile_dim0
        (int)(tile_h & 0xFFFFu),                            // tile_dim1 | tile_dim2=0
        (int)row_stride_elts,                               // tensor_dim0_stride[31:0]
        0, 0
    };
    i32x4 z4 = {0, 0, 0, 0};
    i32x8 z8 = {0, 0, 0, 0, 0, 0, 0, 0};
    __builtin_amdgcn_tensor_load_to_lds(g0, g1, z4, z4, z8, 0);
}

// ---------------- LayerNorm: one 256-thread block per row of 768 ----------------
__global__ __launch_bounds__(256) void ln_kernel(const float* __restrict__ x,
                                                 const float* __restrict__ g,
                                                 const float* __restrict__ b,
                                                 float* __restrict__ out) {
    __shared__ float red[256];
    const int row = blockIdx.x;
    const int tid = threadIdx.x;
    const float* xr = x + (size_t)row * DIM;
    float v0 = xr[tid], v1 = xr[tid + 256], v2 = xr[tid + 512];
    red[tid] = v0 + v1 + v2;
    __syncthreads();
    for (int off = 128; off > 0; off >>= 1) {
        if (tid < off) red[tid] += red[tid + off];
        __syncthreads();
    }
    const float mu = red[0] * (1.0f / DIM);
    __syncthreads();
    const float d0 = v0 - mu, d1 = v1 - mu, d2 = v2 - mu;
    red[tid] = d0 * d0 + d1 * d1 + d2 * d2;
    __syncthreads();
    for (int off = 128; off > 0; off >>= 1) {
        if (tid < off) red[tid] += red[tid + off];
        __syncthreads();
    }
    const float rstd = rsqrtf(red[0] * (1.0f / DIM) + 1e-5f);
    float* o = out + (size_t)row * DIM;
    o[tid]       = d0 * rstd * g[tid]       + b[tid];
    o[tid + 256] = d1 * rstd * g[tid + 256] + b[tid + 256];
    o[tid + 512] = d2 * rstd * g[tid + 512] + b[tid + 512];
}

// ------------- Depthwise 3x3 conv over (b, 768, 14, 14) laid out as h(b,n,c) -------------
__global__ __launch_bounds__(256) void dwconv_kernel(const float* __restrict__ h,
                                                     const float* __restrict__ w,    // (DIM,1,3,3)
                                                     const float* __restrict__ bias, // (DIM)
                                                     float* __restrict__ out) {
    const int n  = blockIdx.x;   // token 0..195
    const int bb = blockIdx.y;   // batch
    const int y  = n / 14, xp = n % 14;
    const float* hb = h   + (size_t)bb * NTOK * DIM;
    float*       ob = out + (size_t)bb * NTOK * DIM;
    for (int j = 0; j < 3; ++j) {
        const int ch = threadIdx.x + j * 256;
        float acc = bias[ch];
        const float* wc = w + ch * 9;
        #pragma unroll
        for (int dy = -1; dy <= 1; ++dy) {
            const int yy = y + dy;
            if (yy < 0 || yy >= 14) continue;
            #pragma unroll
            for (int dx = -1; dx <= 1; ++dx) {
                const int xx = xp + dx;
                if (xx < 0 || xx >= 14) continue;
                acc += wc[(dy + 1) * 3 + (dx + 1)] * hb[(size_t)(yy * 14 + xx) * DIM + ch];
            }
        }
        ob[(size_t)n * DIM + ch] = acc;
    }
}

// ------------- softmax(q*k0, axis=tokens) * k1 + sepe, per (batch, channel) column -------------
__global__ __launch_bounds__(256) void softmax_combine_kernel(
        const float* __restrict__ q,    // (B, N, DIM)
        const float* __restrict__ kv,   // (B, N, 2*DIM)  k0 | k1
        float* __restrict__ a)          // in: sepe, out: a  (B, N, DIM)
{
    __shared__ float red[256];
    const int ch  = blockIdx.x;   // 0..767
    const int bb  = blockIdx.y;   // batch
    const int tid = threadIdx.x;
    const size_t base = (size_t)bb * NTOK;
    float s = -3.0e38f, k1v = 0.0f, sep = 0.0f;
    if (tid < NTOK) {
        const size_t r = base + tid;
        const float qv  = q[r * DIM + ch];
        const float k0v = kv[r * (2 * DIM) + ch];
        k1v = kv[r * (2 * DIM) + DIM + ch];
        sep = a[r * DIM + ch];
        s = qv * k0v;
    }
    red[tid] = s;
    __syncthreads();
    for (int off = 128; off > 0; off >>= 1) {
        if (tid < off) red[tid] = fmaxf(red[tid], red[tid + off]);
        __syncthreads();
    }
    const float mx = red[0];
    __syncthreads();
    const float e = (tid < NTOK) ? expf(s - mx) : 0.0f;
    red[tid] = e;
    __syncthreads();
    for (int off = 128; off > 0; off >>= 1) {
        if (tid < off) red[tid] += red[tid + off];
        __syncthreads();
    }
    const float inv = 1.0f / red[0];
    if (tid < NTOK) {
        const size_t r = base + tid;
        a[r * DIM + ch] = e * inv * k1v + sep;
    }
}

// ---------------- fp32 WMMA GEMM: C(MxN) = A(MxK) @ B(KxN) [+ epilogue] ----------------
// 256 threads = 8 waves; block tile 64x32, K-chunk 32; each wave owns a 16x16 tile.
// TDM double-buffers the LDS tiles (wave 0 issues, TENSORcnt synchronizes), WMMAs
// consume chunk i while the TDM fetches chunk i+1.
constexpr int EPI_NONE = 0;
constexpr int EPI_GELU = 1;
constexpr int EPI_BIAS_RES = 2;
constexpr int EPI_GELU_BIAS = 3;

template <int EPI>
__global__ __launch_bounds__(256) void gemm_wmma_f32(
        const float* __restrict__ A,
        const float* __restrict__ B,
        float* __restrict__ C,
        const float* __restrict__ bias,
        const float* __restrict__ res,
        int K, int N)
{
    constexpr int BM = 64, BN = 32, BK = 32;
    constexpr int LDA = BK + 1;   // 33 floats: TDM pad (+1 DW / 32 DW) kills bank conflicts
    constexpr int LDB = BN + 1;   // 33
    __shared__ float sA[2][BM * LDA];
    __shared__ float sB[2][BK * LDB];

    const int tid  = threadIdx.x;
    const int lane = tid & 31;
    const int wave = tid >> 5;
    const int wr   = wave >> 1;     // 0..3 : 16-row tile within block
    const int wc   = wave & 1;      // 0..1 : 16-col tile within block
    const int m0   = blockIdx.y * BM;
    const int n0   = blockIdx.x * BN;
    const bool issuer = (wave == 0);

    const unsigned aoff0 = (unsigned)(uintptr_t)&sA[0][0];
    const unsigned aoff1 = (unsigned)(uintptr_t)&sA[1][0];
    const unsigned boff0 = (unsigned)(uintptr_t)&sB[0][0];
    const unsigned boff1 = (unsigned)(uintptr_t)&sB[1][0];

    const int mloc = lane & 15;          // M (A) or N (B) within tile
    const int kh   = (lane >> 4) * 2;    // lanes 16..31 hold the K=2,3 pair

    v8f acc = {};
    const int nchunks = K / BK;

    if (issuer) {
        tdm_load_tile_f32(aoff0, A + (size_t)m0 * K, (unsigned)K, BK, BM);
        tdm_load_tile_f32(boff0, B + (size_t)n0,     (unsigned)N, BN, BK);
    }

    for (int c = 0; c < nchunks; ++c) {
        if (issuer) __builtin_amdgcn_s_wait_tensorcnt(0);
        __syncthreads();   // chunk c resident in LDS for all waves
        if (issuer && (c + 1) < nchunks) {
            const int k1 = (c + 1) * BK;
            tdm_load_tile_f32((c & 1) ? aoff0 : aoff1,
                              A + (size_t)m0 * K + k1, (unsigned)K, BK, BM);
            tdm_load_tile_f32((c & 1) ? boff0 : boff1,
                              B + (size_t)k1 * N + n0, (unsigned)N, BN, BK);
        }
        const float* cA = &sA[c & 1][0];
        const float* cB = &sB[c & 1][0];
        #pragma unroll
        for (int kk = 0; kk < BK; kk += 4) {
            v2f av, bv;
            const float* ap = &cA[(wr * 16 + mloc) * LDA + kk + kh];
            av.x = ap[0];
            av.y = ap[1];
            const float* bp = &cB[(kk + kh) * LDB + wc * 16 + mloc];
            bv.x = bp[0];
            bv.y = bp[LDB];
            acc = __builtin_amdgcn_wmma_f32_16x16x4_f32(
                    /*neg_a=*/false, av, /*neg_b=*/false, bv,
                    /*c_mod=*/(short)0, acc, /*reuse_a=*/false, /*reuse_b=*/false);
        }
        __syncthreads();   // everyone done with buffer (c&1) before it is refilled
    }

    // C/D layout: VGPR i -> row (i + 8*(lane>=16)), col = lane&15
    const int row0 = m0 + wr * 16 + ((lane >> 4) * 8);
    const int col  = n0 + wc * 16 + (lane & 15);
    #pragma unroll
    for (int i = 0; i < 8; ++i) {
        float v = acc[i];
        const size_t idx = (size_t)(row0 + i) * N + col;
        if (EPI == EPI_BIAS_RES || EPI == EPI_GELU_BIAS) v += bias[col];
        if (EPI == EPI_GELU || EPI == EPI_GELU_BIAS) v = gelu_exact(v);
        if (EPI == EPI_BIAS_RES) v += res[idx];
        C[idx] = v;
    }
}

extern "C" void kernel_launch(void* const* d_in, const int* in_sizes, int n_in,
                              void* d_out, int out_size, void* d_ws, size_t ws_size,
                              hipStream_t stream) {
    const float* x     = (const float*)d_in[0];
    const float* ln1g  = (const float*)d_in[1];
    const float* ln1b  = (const float*)d_in[2];
    const float* qW1   = (const float*)d_in[3];
    const float* qW2   = (const float*)d_in[4];
    const float* kvW   = (const float*)d_in[5];
    const float* posW  = (const float*)d_in[6];
    const float* posB  = (const float*)d_in[7];
    const float* projW = (const float*)d_in[8];
    const float* projB = (const float*)d_in[9];
    const float* ln2g  = (const float*)d_in[10];
    const float* ln2b  = (const float*)d_in[11];
    const float* fc1W  = (const float*)d_in[12];
    const float* fc1b  = (const float*)d_in[13];
    const float* fc2W  = (const float*)d_in[14];
    const float* fc2b  = (const float*)d_in[15];
    float* out = (float*)d_out;
    float* ws  = (float*)d_ws;

    // Workspace layout (floats), with reuse:
    float* h  = ws;                             // ROWS*DIM       (LN1 output)
    float* kv = h  + (size_t)ROWS * DIM;        // ROWS*2*DIM     (k0|k1)
    float* tm = kv + (size_t)ROWS * 2 * DIM;    // ROWS*MLPD      (t=gelu(h@qW1), later m1)
    float* qb = tm + (size_t)ROWS * MLPD;       // ROWS*DIM       (q, later h2)
    float* sp = qb + (size_t)ROWS * DIM;        // ROWS*DIM       (sepe, overwritten by a)

    // 1) h = LN1(x)
    ln_kernel<<<ROWS, 256, 0, stream>>>(x, ln1g, ln1b, h);
    // 2) sepe = dwconv3x3(h) + posB
    dwconv_kernel<<<dim3(NTOK, BATCH), 256, 0, stream>>>(h, posW, posB, sp);
    // 3) kv = h @ kvW                        (12544 x 768 x 1536)
    gemm_wmma_f32<EPI_NONE><<<dim3((2 * DIM) / 32, ROWS / 64), 256, 0, stream>>>(
        h, kvW, kv, nullptr, nullptr, DIM, 2 * DIM);
    // 4) t = gelu(h @ qW1)                   (12544 x 768 x 64)
    gemm_wmma_f32<EPI_GELU><<<dim3(HID / 32, ROWS / 64), 256, 0, stream>>>(
        h, qW1, tm, nullptr, nullptr, DIM, HID);
    // 5) q = t @ qW2                         (12544 x 64 x 768)
    gemm_wmma_f32<EPI_NONE><<<dim3(DIM / 32, ROWS / 64), 256, 0, stream>>>(
        tm, qW2, qb, nullptr, nullptr, HID, DIM);
    // 6) a = softmax(q*k0, axis=tokens)*k1 + sepe   (in-place into sp)
    softmax_combine_kernel<<<dim3(DIM, BATCH), 256, 0, stream>>>(qb, kv, sp);
    // 7) x1 = x + a @ projW + projB  -> d_out
    gemm_wmma_f32<EPI_BIAS_RES><<<dim3(DIM / 32, ROWS / 64), 256, 0, stream>>>(
        sp, projW, out, projB, x, DIM, DIM);
    // 8) h2 = LN2(x1)
    ln_kernel<<<ROWS, 256, 0, stream>>>(out, ln2g, ln2b, qb);
    // 9) m1 = gelu(h2 @ fc1W + fc1b)         (12544 x 768 x 3072)
    gemm_wmma_f32<EPI_GELU_BIAS><<<dim3(MLPD / 32, ROWS / 64), 256, 0, stream>>>(
        qb, fc1W, tm, fc1b, nullptr, DIM, MLPD);
    // 10) out = x1 + m1 @ fc2W + fc2b        (12544 x 3072 x 768)
    gemm_wmma_f32<EPI_BIAS_RES><<<dim3(DIM / 32, ROWS / 64), 256, 0, stream>>>(
        tm, fc2W, out, fc2b, out, MLPD, DIM);
}